// DetectorLossFn_46265387713197
// MI455X (gfx1250) — compile-verified
//
#include <hip/hip_runtime.h>
#include <cstdint>

// ---------------------------------------------------------------------------
// DetectorLoss for MI455X (gfx1250).
//   d_in[0] = pred_boxes (16, 32768, 5) f32
//   d_in[1] = pred_cls   (16, 32768, 16) f32
//   d_in[2] = target     (16, 128, 5) f32
//   d_out   = 7 f32: (loss, loss_cls, loss_x, loss_y, loss_w, loss_h, loss_conf)
// Workspace layout (d_ws):
//   [0      .. 16384)  : u64 best[N*M]   packed (mapped_iou<<32)|(~k)  (argmax)
//   [16384  .. 16832)  : f32 partial[16][7] per-block loss partial sums
// ---------------------------------------------------------------------------

namespace {
constexpr int N = 16;
constexpr int K = 32768;
constexpr int C = 16;
constexpr int M = 128;
constexpr int SPLIT = 16;          // K-split: blocks per image in phase 1
constexpr int KC = K / SPLIT;      // 2048 predictions per block
}

typedef __attribute__((ext_vector_type(2))) float v2f;
typedef __attribute__((ext_vector_type(8))) float v8f;

// Monotone float -> u32 map (preserves total order incl. negatives).
__device__ __forceinline__ unsigned order_key(float f) {
  unsigned u = __float_as_uint(f);
  return (u & 0x80000000u) ? ~u : (u | 0x80000000u);
}

// ---------------------------------------------------------------------------
// Phase 0: zero the argmax slots and partial-sum area (harness poisons d_ws).
// ---------------------------------------------------------------------------
__global__ void init_ws_kernel(unsigned long long* __restrict__ best,
                               float* __restrict__ partial) {
  int i = blockIdx.x * blockDim.x + threadIdx.x;
  if (i < N * M) best[i] = 0ull;          // 0 is below every real packed key
  if (i < N * 7) partial[i] = 0.0f;
}

// ---------------------------------------------------------------------------
// Phase 1: IoU argmax. Grid = N*SPLIT blocks of 256 threads.
// Block (n, c): stage pred_boxes[n, c*KC .. c*KC+KC) (raw 5-f32 rows, 40960 B)
// into LDS with CDNA5 async global->LDS copies, then 2 threads per target scan
// 1024 predictions each with broadcast LDS reads; combine via u64 atomicMax.
// ---------------------------------------------------------------------------
__global__ __launch_bounds__(256) void iou_argmax_kernel(
    const float* __restrict__ pred_boxes,
    const float* __restrict__ target,
    unsigned long long* __restrict__ best) {
  __shared__ __align__(16) float tile[KC * 5];          // 40960 bytes
  __shared__ unsigned long long cand[M];

  const int n = blockIdx.x / SPLIT;
  const int c = blockIdx.x % SPLIT;
  const int t = threadIdx.x;

  // ---- async copy chunk into LDS: 2560 x 16B, 10 per thread, all aligned ---
  const float* src = pred_boxes + ((size_t)n * K + (size_t)c * KC) * 5;
  unsigned lds_base = (unsigned)(uintptr_t)(&tile[0]);
  unsigned long long gbase = (unsigned long long)(uintptr_t)src;
#pragma unroll
  for (int i = 0; i < 10; ++i) {
    int chunk = i * 256 + t;                             // 0..2559
    unsigned la = lds_base + (unsigned)chunk * 16u;
    unsigned long long ga = gbase + (unsigned long long)chunk * 16ull;
    asm volatile("global_load_async_to_lds_b128 %0, %1, off"
                 :: "v"(la), "v"(ga) : "memory");
  }
  asm volatile("s_wait_asynccnt 0" ::: "memory");
  __syncthreads();

  // ---- per-thread target setup: 2 threads per target, halves of the chunk --
  const int m0 = t & (M - 1);
  const int half = t >> 7;                               // 0 or 1
  const float* trow = target + ((size_t)n * M + m0) * 5;
  const float t1 = trow[1], t2 = trow[2], t3 = trow[3], t4 = trow[4];
  const float a2 = (t3 - t1 + 1.0f) * (t4 - t2 + 1.0f);

  unsigned long long bestpack = 0ull;
  const int p0 = half * (KC / 2);
#pragma unroll 4
  for (int i = 0; i < KC / 2; ++i) {
    const int p = p0 + i;
    const float* q = &tile[p * 5];                       // wave-broadcast read
    const float x = q[0], y = q[1], w = q[2], h = q[3];
    const float px2 = w + x, py2 = h + y;
    const float a1 = (w + 1.0f) * (h + 1.0f);
    const float ix1 = fmaxf(x, t1), iy1 = fmaxf(y, t2);
    const float ix2 = fminf(px2, t3), iy2 = fminf(py2, t4);
    const float iw = fmaxf(ix2 - ix1 + 1.0f, 0.0f);
    const float ih = fmaxf(iy2 - iy1 + 1.0f, 0.0f);
    const float inter = iw * ih;
    const float iou = inter / (a1 + a2 - inter + 1e-16f);
    const unsigned kglob = (unsigned)(c * KC + p);
    const unsigned long long pack =
        ((unsigned long long)order_key(iou) << 32) | (0xFFFFFFFFu - kglob);
    bestpack = (pack > bestpack) ? pack : bestpack;      // ties impossible
  }

  if (half == 1) cand[m0] = bestpack;
  __syncthreads();
  if (half == 0) {
    unsigned long long other = cand[m0];
    if (other > bestpack) bestpack = other;
    atomicMax(&best[(size_t)n * M + m0], bestpack);
  }
}

// ---------------------------------------------------------------------------
// Phase 2: per-target loss terms + deterministic block reduction.
// Grid = N blocks of 256 threads (8 waves). Wave w handles targets w*16..+15;
// lanes L and L+16 both own row (L&15) so the 16x4 f32 WMMA A-layout is fed
// directly: lanes 0-15 hold K=0,1 (VGPR0/1), lanes 16-31 hold K=2,3.
// B = ones (layout-invariant) -> D[i][*] = sum_k exp(logit[i][k]-max[i]).
// ---------------------------------------------------------------------------
__global__ __launch_bounds__(256) void loss_kernel(
    const float* __restrict__ pred_boxes,
    const float* __restrict__ pred_cls,
    const float* __restrict__ target,
    const unsigned long long* __restrict__ best,
    float* __restrict__ partial) {
  __shared__ float wsum[8][7];

  const int n = blockIdx.x;
  const int t = threadIdx.x;
  const int lane = t & 31;
  const int wave = t >> 5;
  const int r = lane & 15;
  const int m = wave * 16 + r;

  // conf_idx == 0 everywhere: best_conf = sigmoid(pred_boxes[n,0,4])
  const float pconf = pred_boxes[(size_t)n * K * 5 + 4];
  const float conf = 1.0f / (1.0f + expf(-pconf));
  const float label = (conf > 0.5f) ? 1.0f : 0.0f;
  const float bce = -(label * logf(conf) + (1.0f - label) * logf(1.0f - conf));

  const float* trow = target + ((size_t)n * M + m) * 5;
  const float t0 = trow[0], t1 = trow[1], t2 = trow[2], t3 = trow[3],
              t4 = trow[4];
  const float maskf = ((t0 + t1 + t2 + t3 + t4) != 0.0f) ? 1.0f : 0.0f;

  const unsigned long long pk = best[(size_t)n * M + m];
  int kb = (int)(0xFFFFFFFFu - (unsigned)(pk & 0xFFFFFFFFull));
  kb = kb < 0 ? 0 : (kb >= K ? K - 1 : kb);

  const float* pbrow = pred_boxes + ((size_t)n * K + kb) * 5;
  const float bx = pbrow[0], by = pbrow[1], bw = pbrow[2], bh = pbrow[3];

  // 16 class logits of the best row (64B aligned).
  float xc[16];
  {
    const float4* crow =
        (const float4*)(pred_cls + ((size_t)n * K + kb) * C);
#pragma unroll
    for (int i = 0; i < 4; ++i) {
      float4 v = crow[i];
      xc[4 * i + 0] = v.x; xc[4 * i + 1] = v.y;
      xc[4 * i + 2] = v.z; xc[4 * i + 3] = v.w;
    }
  }
  float mx = xc[0];
#pragma unroll
  for (int i = 1; i < 16; ++i) mx = fmaxf(mx, xc[i]);

  // ---- sum_c exp(xc - mx) via V_WMMA_F32_16X16X4_F32, 4 accumulating steps -
  float se;
#if __has_builtin(__builtin_amdgcn_wmma_f32_16x16x4_f32)
  {
    v8f acc = {};
    v2f bones; bones.x = 1.0f; bones.y = 1.0f;
    const int sub = (lane & 16) ? 2 : 0;   // lanes 16-31 feed K=2,3
#pragma unroll
    for (int kc = 0; kc < 4; ++kc) {
      v2f a;
      a.x = expf(xc[4 * kc + sub + 0] - mx);
      a.y = expf(xc[4 * kc + sub + 1] - mx);
      acc = __builtin_amdgcn_wmma_f32_16x16x4_f32(
          false, a, false, bones, (short)0, acc, false, false);
    }
    // D layout: VGPR v, lanes0-15 -> M=v, lanes16-31 -> M=8+v.
    // Row L lives at (vgpr L&7, lane L) for L<8 and (vgpr L&7, lane L+8) else.
    float se_part = acc[lane & 7];
    int srcl = (lane & 8) ? (lane + 8) : lane;
    se = __shfl(se_part, srcl & 31, 32);
  }
#else
  {
    se = 0.0f;
#pragma unroll
    for (int i = 0; i < 16; ++i) se += expf(xc[i] - mx);
  }
#endif

  int tc = (int)t0;
  tc = tc < 0 ? 0 : (tc > 15 ? 15 : tc);
  const float ce = logf(se) + mx - xc[tc];        // -log_softmax[tc]

  const float dx = bx - t1;
  const float dy = by - t2;
  const float dw = bw - (t3 - t1);
  const float dh = bh - (t4 - t2);

  // Only lanes 0-15 contribute (lanes 16-31 are layout duplicates).
  const float g = (lane < 16) ? maskf : 0.0f;
  float v[7] = {g * ce, g * dx * dx, g * dy * dy,
                g * dw * dw, g * dh * dh, g * bce, g};

  // wave32 xor-reduction, then fixed-order sum over 8 waves.
#pragma unroll
  for (int j = 0; j < 7; ++j) {
#pragma unroll
    for (int off = 16; off > 0; off >>= 1)
      v[j] += __shfl_xor(v[j], off, 32);
  }
  if (lane == 0) {
#pragma unroll
    for (int j = 0; j < 7; ++j) wsum[wave][j] = v[j];
  }
  __syncthreads();
  if (t < 7) {
    float s = 0.0f;
#pragma unroll
    for (int w = 0; w < 8; ++w) s += wsum[w][t];
    partial[n * 7 + t] = s;
  }
}

// ---------------------------------------------------------------------------
// Phase 3: fixed-order final reduction over 16 block partials -> 7 outputs.
// ---------------------------------------------------------------------------
__global__ void finalize_kernel(const float* __restrict__ partial,
                                float* __restrict__ out) {
  const int t = threadIdx.x;        // launched with 32 threads
  float s = 0.0f;
  if (t < 7) {
    for (int b = 0; b < N; ++b) s += partial[b * 7 + t];
  }
  const float denom = __shfl(s, 6, 32);
  float val = (t < 6) ? (s / denom) : 0.0f;
  if (t < 6) out[1 + t] = val;      // cls, x, y, w, h, conf
  float tot = val;
#pragma unroll
  for (int off = 16; off > 0; off >>= 1) tot += __shfl_xor(tot, off, 32);
  if (t == 0) out[0] = tot;         // total loss
}

// ---------------------------------------------------------------------------
extern "C" void kernel_launch(void* const* d_in, const int* in_sizes, int n_in,
                              void* d_out, int out_size, void* d_ws,
                              size_t ws_size, hipStream_t stream) {
  (void)in_sizes; (void)n_in; (void)out_size; (void)ws_size;
  const float* pred_boxes = (const float*)d_in[0];
  const float* pred_cls   = (const float*)d_in[1];
  const float* target     = (const float*)d_in[2];
  float* out = (float*)d_out;

  unsigned long long* best = (unsigned long long*)d_ws;
  float* partial = (float*)((char*)d_ws + (size_t)N * M * sizeof(unsigned long long));

  init_ws_kernel<<<(N * M + 255) / 256, 256, 0, stream>>>(best, partial);
  iou_argmax_kernel<<<N * SPLIT, 256, 0, stream>>>(pred_boxes, target, best);
  loss_kernel<<<N, 256, 0, stream>>>(pred_boxes, pred_cls, target, best, partial);
  finalize_kernel<<<1, 32, 0, stream>>>(partial, out);
}